// ElectronHoleBasisAssembly_TensorProduct_2559800508920
// MI455X (gfx1250) — compile-verified
//
#include <hip/hip_runtime.h>

// Problem constants from the reference: B=8, NK=128, NC=8, NV=8, D=32.
#define NC 8
#define NV 8
#define DIM 32

typedef __attribute__((ext_vector_type(2))) float v2f;
typedef __attribute__((ext_vector_type(8))) float v8f;

// One wave computes one (b,k,v,c) 32x32 outer-product tile:
//   out[i][j] = x1[b,k,c,i] * x2[b,k,v,j]
// via four V_WMMA_F32_16X16X4_F32 rank-1 updates (K=0 column of A holds a,
// every K row of B holds b, so D = a[m]*b[n]).
__global__ __launch_bounds__(256)
void ElectronHoleBasisAssembly_TensorProduct_kernel(
    const float* __restrict__ x1,   // (B*NK, NC, D)
    const float* __restrict__ x2,   // (B*NK, NV, D)
    float* __restrict__ out)        // (B*NK, NV, NC, D, D)
{
  const int lane = threadIdx.x & 31;
  const int wave = threadIdx.x >> 5;
  const int tile = (blockIdx.x << 3) + wave;   // grid sized exactly: no guard,
                                               // EXEC stays all-1s for WMMA.

  const int vc = tile & 63;        // (v,c) within the (b,k) pair
  const int bk = tile >> 6;        // flattened (b,k)
  const int v  = vc >> 3;
  const int c  = vc & 7;

  const float* a_row = x1 + (size_t)(bk * NC + c) * DIM;  // 32 floats
  const float* b_row = x2 + (size_t)(bk * NV + v) * DIM;  // 32 floats

  const int  l16 = lane & 15;
  const bool lo  = lane < 16;

  // A fragment (16x4, f32): lanes 0-15 hold M=lane; component.x lanes0-15 = K0.
  // Only K=0 is nonzero -> rank-1 product.
  const float a0v = lo ? a_row[l16]      : 0.0f;   // i-half 0: i = lane
  const float a1v = lo ? a_row[16 + l16] : 0.0f;   // i-half 1: i = 16 + lane
  v2f A0; A0.x = a0v; A0.y = 0.0f;
  v2f A1; A1.x = a1v; A1.y = 0.0f;

  // B fragment (4x16, f32): replicate b into every K row (both components,
  // both lane halves) -> D[m,n] = a[m]*b[n] regardless of K-row placement.
  const float b0v = b_row[l16];        // j-half 0
  const float b1v = b_row[16 + l16];   // j-half 1
  v2f B0; B0.x = b0v; B0.y = b0v;
  v2f B1; B1.x = b1v; B1.y = b1v;

  v8f zero = {};
  // 8 args: (neg_a, A, neg_b, B, c_mod, C, reuse_a, reuse_b)
  v8f d00 = __builtin_amdgcn_wmma_f32_16x16x4_f32(false, A0, false, B0, (short)0, zero, false, false);
  v8f d01 = __builtin_amdgcn_wmma_f32_16x16x4_f32(false, A0, false, B1, (short)0, zero, false, false);
  v8f d10 = __builtin_amdgcn_wmma_f32_16x16x4_f32(false, A1, false, B0, (short)0, zero, false, false);
  v8f d11 = __builtin_amdgcn_wmma_f32_16x16x4_f32(false, A1, false, B1, (short)0, zero, false, false);

  // D layout (16x16 f32): VGPR r, lanes 0-15 -> row r, col lane;
  // lanes 16-31 -> row r+8, col lane-16. Store straight from accumulators:
  // each component store is two 64B-contiguous lane groups (fully coalesced).
  float* o = out + (size_t)tile * (DIM * DIM);
  const int mbase = lo ? 0 : 8;
  const int n     = l16;
#pragma unroll
  for (int r = 0; r < 8; ++r) {
    const int m = mbase + r;
    o[(size_t)m        * DIM + n]      = d00[r];
    o[(size_t)m        * DIM + 16 + n] = d01[r];
    o[(size_t)(16 + m) * DIM + n]      = d10[r];
    o[(size_t)(16 + m) * DIM + 16 + n] = d11[r];
  }
}

extern "C" void kernel_launch(void* const* d_in, const int* in_sizes, int n_in,
                              void* d_out, int out_size, void* d_ws, size_t ws_size,
                              hipStream_t stream) {
  (void)in_sizes; (void)n_in; (void)d_ws; (void)ws_size;
  const float* x1  = (const float*)d_in[0];
  const float* x2  = (const float*)d_in[1];
  float*       out = (float*)d_out;

  // out_size = B*NK*NV*NC*D*D = 67,108,864 -> 65,536 tiles of 32x32.
  const int nTiles = out_size / (DIM * DIM);
  const int blocks = nTiles / 8;            // 8 waves (tiles) per 256-thread block
  if (blocks <= 0) return;

  hipLaunchKernelGGL(ElectronHoleBasisAssembly_TensorProduct_kernel,
                     dim3(blocks), dim3(256), 0, stream, x1, x2, out);
}